// QCNN_torch_40673340293399
// MI455X (gfx1250) — compile-verified
//
#include <hip/hip_runtime.h>

// ---------------------------------------------------------------------------
// QCNN fused kernel for MI455X (gfx1250, wave32, WMMA).
//
// Roofline: 67MB (x) + 10.5MB (fc_w) at 23.3 TB/s ~= 3.3us. Compute ~2.8
// GFLOP is free; the 16-state -> 16-row transform of 16 patches at a time is
// done with exact-precision chained v_wmma_f32_16x16x4_f32 (4 per tile).
// Workspace use: 1KB for the 16x16 M matrix + 640*2048*4 = 5.25MB partials.
// ---------------------------------------------------------------------------

typedef __attribute__((ext_vector_type(2))) float v2f;
typedef __attribute__((ext_vector_type(8))) float v8f;

struct cpx { float re, im; };
__device__ inline cpx cmul(cpx a, cpx b) {
    return {a.re * b.re - a.im * b.im, a.re * b.im + a.im * b.re};
}
__device__ inline cpx cadd(cpx a, cpx b) { return {a.re + b.re, a.im + b.im}; }

// Controlled gate on 2 qubits (q0 = MSB = target, q1 = LSB = control):
// G = CRX(trx) @ CRZ(trz). CRZ is diagonal.
__device__ void gate2(float trz, float trx, cpx G[4][4]) {
    float hz = 0.5f * trz, hx = 0.5f * trx;
    float cz = cosf(hz), sz = sinf(hz);
    float cx = cosf(hx), sx = sinf(hx);
    for (int a = 0; a < 4; ++a)
        for (int b = 0; b < 4; ++b) {
            cpx v = {0.f, 0.f};
            if ((a & 1) == (b & 1)) {            // control bits match
                int ta = a >> 1, tb = b >> 1;    // target bits
                if ((a & 1) == 0) { if (ta == tb) v = {1.f, 0.f}; }
                else v = (ta == tb) ? cpx{cx, 0.f} : cpx{0.f, -sx};
            }
            cpx dz = (b & 1) ? ((b >> 1) ? cpx{cz, sz} : cpx{cz, -sz})
                             : cpx{1.f, 0.f};
            G[a][b] = cmul(v, dz);
        }
}

// Row 0 of C3 = ctrl(RX,3,2,0) @ ctrl(RZ,3,2,0) on 3 qubits
// (q0 = MSB = target, q2 = LSB = control, q1 must match).
__device__ void gate3_row0(float trz, float trx, cpx R[8]) {
    float hz = 0.5f * trz, hx = 0.5f * trx;
    float cz = cosf(hz), sz = sinf(hz);
    float cx = cosf(hx), sx = sinf(hx);
    const int a = 0;
    for (int b = 0; b < 8; ++b) {
        cpx v = {0.f, 0.f};
        if ((a & 1) == (b & 1) && (((a >> 1) & 1) == ((b >> 1) & 1))) {
            int ta = a >> 2, tb = b >> 2;
            if ((a & 1) == 0) { if (ta == tb) v = {1.f, 0.f}; }
            else v = (ta == tb) ? cpx{cx, 0.f} : cpx{0.f, -sx};
        }
        cpx dz = (b & 1) ? ((b >> 2) ? cpx{cz, sz} : cpx{cz, -sz})
                         : cpx{1.f, 0.f};
        R[b] = cmul(v, dz);
    }
}

// Build M[16][16]: for channel o, rows 4o+0/1 = Re/Im of U2 row0,
// rows 4o+2/3 = Re/Im of U2 row1.  U2[r][j] = sum_m C3row0[m]*L1[2m+r][j],
// L1[k][j] = G01[k>>2][j>>2] * G23[k&3][j&3].
__global__ void build_M_kernel(const float* __restrict__ qp,
                               float* __restrict__ M) {
    int o = threadIdx.x;
    if (o >= 4) return;
    const float* p = qp + o * 6;
    cpx G01[4][4], G23[4][4], C3r[8];
    gate2(p[0], p[1], G01);
    gate2(p[2], p[3], G23);
    gate3_row0(p[4], p[5], C3r);
    for (int r = 0; r < 2; ++r)
        for (int j = 0; j < 16; ++j) {
            cpx s = {0.f, 0.f};
            for (int m = 0; m < 8; ++m) {
                int k = 2 * m + r;
                s = cadd(s, cmul(C3r[m],
                                 cmul(G01[k >> 2][j >> 2], G23[k & 3][j & 3])));
            }
            M[(4 * o + 2 * r + 0) * 16 + j] = s.re;
            M[(4 * o + 2 * r + 1) * 16 + j] = s.im;
        }
}

__device__ inline v8f wmma4(v2f a, v2f b, v8f c) {
    // D = A(16x4 f32) * B(4x16 f32) + C, exact f32 accumulation.
    return __builtin_amdgcn_wmma_f32_16x16x4_f32(
        false, a, false, b, (short)0, c, false, false);
}

__device__ inline void load_patch4(const float* __restrict__ x, int b, int hb,
                                   int xcol, float2 v[4]) {
    const float* q0 = x + (size_t)b * 262144u + (size_t)(4 * hb) * 512 + xcol;
    v[0] = *(const float2*)(q0);          // h row 2hb,  x row 4hb   (top)
    v[1] = *(const float2*)(q0 + 512);    //             x row 4hb+1 (bottom)
    v[2] = *(const float2*)(q0 + 1024);   // h row 2hb+1
    v[3] = *(const float2*)(q0 + 1536);
}

// One wave per (batch-group, h-pair, w-tile). grid = 4*128*16 = 8192 waves.
__global__ __launch_bounds__(32) void qconv_main(
    const float* __restrict__ x, const float* __restrict__ fcw,
    const float* __restrict__ M, float* __restrict__ partial) {
    const int bg   = blockIdx.x >> 11;    // 0..3  (16 batches each)
    const int pos  = blockIdx.x & 2047;   // 0..2047
    const int hb   = pos >> 4;            // h rows 2hb, 2hb+1
    const int wt   = pos & 15;            // 16 patches in w
    const int lane = threadIdx.x;
    const int p    = lane & 15;           // patch within tile (= WMMA N)
    const bool hi  = lane >= 16;

    // A operand regs: 16x4 f32 layout -> V0: K={0|2}, V1: K={1|3} per half.
    v2f A[4];
#pragma unroll
    for (int c = 0; c < 4; ++c) {
        int kb = 4 * c + (hi ? 2 : 0);
        A[c].x = M[p * 16 + kb];
        A[c].y = M[p * 16 + kb + 1];
    }

    // fc_w for this lane's patch & its two channels, both h rows: 40 regs.
    const int w   = wt * 16 + p;
    const int olo = hi ? 2 : 0;
    float wA[2][10], wB[2][10], msk[2];
#pragma unroll
    for (int hh = 0; hh < 2; ++hh) {
        int h = 2 * hb + hh;
        int idx = h * 256 + w;
#pragma unroll
        for (int c = 0; c < 10; ++c) {
            wA[hh][c] = fcw[c * 262144 + olo * 65536 + idx];
            wB[hh][c] = fcw[c * 262144 + (olo + 1) * 65536 + idx];
        }
        msk[hh] = (h < 255 && w < 255) ? 1.f : 0.f;  // ref skips last row/col
    }

    const int xcol = wt * 32 + 2 * p;
    const int b0   = bg * 16;

    float2 cur[4], nxt[4];
    load_patch4(x, b0, hb, xcol, cur);

    for (int i = 0; i < 16; ++i) {
        int b = b0 + i;
        if (i < 15) load_patch4(x, b + 1, hb, xcol, nxt);  // prefetch

        float acc[10];
#pragma unroll
        for (int c = 0; c < 10; ++c) acc[c] = 0.f;

#pragma unroll
        for (int hh = 0; hh < 2; ++hh) {
            float2 ra = cur[2 * hh], rb = cur[2 * hh + 1];
            float c0 = __cosf(0.5f * ra.x), s0 = __sinf(0.5f * ra.x);
            float c1 = __cosf(0.5f * ra.y), s1 = __sinf(0.5f * ra.y);
            float c2 = __cosf(0.5f * rb.x), s2 = __sinf(0.5f * rb.x);
            float c3 = __cosf(0.5f * rb.y), s3 = __sinf(0.5f * rb.y);
            float t2 = hi ? s2 : c2;       // qubit-2 split across lane halves
            float p00 = c0 * c1, p01 = c0 * s1, p10 = s0 * c1, p11 = s0 * s1;

            v8f d = {};
            v2f bb;
            float f;
            f = p00 * t2; bb.x = f * c3; bb.y = f * s3; d = wmma4(A[0], bb, d);
            f = p01 * t2; bb.x = f * c3; bb.y = f * s3; d = wmma4(A[1], bb, d);
            f = p10 * t2; bb.x = f * c3; bb.y = f * s3; d = wmma4(A[2], bb, d);
            f = p11 * t2; bb.x = f * c3; bb.y = f * s3; d = wmma4(A[3], bb, d);

            // rows 4o..4o+3 = Re0,Im0,Re1,Im1 -> |amp0|^2 - |amp1|^2
            float e0 = d[0]*d[0] + d[1]*d[1] - d[2]*d[2] - d[3]*d[3];
            float e1 = d[4]*d[4] + d[5]*d[5] - d[6]*d[6] - d[7]*d[7];
            e0 = fmaxf(e0, 0.f) * msk[hh];
            e1 = fmaxf(e1, 0.f) * msk[hh];
#pragma unroll
            for (int c = 0; c < 10; ++c)
                acc[c] = __builtin_fmaf(e0, wA[hh][c],
                         __builtin_fmaf(e1, wB[hh][c], acc[c]));
        }

        // Wave reduction over 32 lanes (all patches + all 4 channels).
#pragma unroll
        for (int c = 0; c < 10; ++c) {
#pragma unroll
            for (int ofs = 16; ofs > 0; ofs >>= 1)
                acc[c] += __shfl_xor(acc[c], ofs, 32);
        }
        if (lane < 10) {
            float v = acc[0];
#pragma unroll
            for (int c = 1; c < 10; ++c) v = (lane == c) ? acc[c] : v;
            partial[(size_t)(b * 10 + lane) * 2048 + pos] = v;
        }

#pragma unroll
        for (int k = 0; k < 4; ++k) cur[k] = nxt[k];
    }
}

// 640 rows x 2048 partials -> out[b*10+cls] = sum + fc_b[cls].
__global__ __launch_bounds__(256) void reduce_kernel(
    const float* __restrict__ partial, const float* __restrict__ fcb,
    float* __restrict__ out) {
    __shared__ float sm[256];
    int row = blockIdx.x, tid = threadIdx.x;
    const float* src = partial + (size_t)row * 2048;
    float s = 0.f;
    for (int u = tid; u < 2048; u += 256) s += src[u];
    sm[tid] = s;
    __syncthreads();
    for (int st = 128; st > 0; st >>= 1) {
        if (tid < st) sm[tid] += sm[tid + st];
        __syncthreads();
    }
    if (tid == 0) out[row] = sm[0] + fcb[row % 10];
}

extern "C" void kernel_launch(void* const* d_in, const int* in_sizes, int n_in,
                              void* d_out, int out_size, void* d_ws,
                              size_t ws_size, hipStream_t stream) {
    (void)in_sizes; (void)n_in; (void)out_size; (void)ws_size;
    const float* x   = (const float*)d_in[0];   // [64,1,512,512]
    const float* qp  = (const float*)d_in[1];   // [4,6]
    const float* fcw = (const float*)d_in[2];   // [10,262144]
    const float* fcb = (const float*)d_in[3];   // [10]
    float* M       = (float*)d_ws;              // 256 floats
    float* partial = M + 256;                   // 640*2048 floats (5.25MB)

    build_M_kernel<<<1, 4, 0, stream>>>(qp, M);
    qconv_main<<<8192, 32, 0, stream>>>(x, fcw, M, partial);
    reduce_kernel<<<640, 256, 0, stream>>>(partial, fcb, (float*)d_out);
}